// AEVComputer_15436112462136
// MI455X (gfx1250) — compile-verified
//
#include <hip/hip_runtime.h>

typedef float v2f __attribute__((ext_vector_type(2)));
typedef float v8f __attribute__((ext_vector_type(8)));

#define A_ATOMS 96
#define NNEI    95
#define NPAD    96
#define NPAIR   4465     // 95*94/2
#define NQ      10
#define FEAT    384      // 64 radial + 320 angular
#define NBATCH  140      // ceil(4465/32), divisible by 4
#define BPW     35       // batches per wave
#define CPW     6        // radial chunks per wave (24/4)

// cos/sin of ShfZ = (z+0.5)*pi/8
__device__ __constant__ float CZ[8] = {
     0.98078528f,  0.83146961f,  0.55557023f,  0.19509032f,
    -0.19509032f, -0.55557023f, -0.83146961f, -0.98078528f };
__device__ __constant__ float SZ[8] = {
     0.19509032f,  0.55557023f,  0.83146961f,  0.98078528f,
     0.98078528f,  0.83146961f,  0.55557023f,  0.19509032f };

__global__ __launch_bounds__(128)
void aev_kernel(const int* __restrict__ species,
                const float* __restrict__ coords,
                float* __restrict__ out) {
    const int atomGlobal = blockIdx.x;            // c*96 + a
    const int c    = atomGlobal / A_ATOMS;
    const int a    = atomGlobal % A_ATOMS;
    const int tid  = threadIdx.x;
    const int lane = tid & 31;
    // wave id is wave-uniform: force it scalar so loop control stays on the SALU
    const int wv   = __builtin_amdgcn_readfirstlane(tid >> 5);   // 0..3
    const int half = lane >> 4;                   // 0: lanes 0-15, 1: lanes 16-31
    const int l16  = lane & 15;
    const int k0   = half * 2;                    // K index of .x fragment element

    __shared__ float4 sVD[NPAD];                  // (vx, vy, vz, d)
    __shared__ float2 sRad[NPAD];                 // (d, fc_r)
    __shared__ float2 sFS[NPAD];                  // (fc_a, bitcast(species))
    __shared__ int    sSp[NPAD];                  // species (radial mask)
    __shared__ float4 sPair[128];                 // staged (ca, sin, dm, fcj)
    __shared__ int    sQ[128];                    // staged species-pair bin
    __shared__ float  sRed[4][FEAT];

    const float cx = coords[(c * A_ATOMS + a) * 3 + 0];
    const float cy = coords[(c * A_ATOMS + a) * 3 + 1];
    const float cz = coords[(c * A_ATOMS + a) * 3 + 2];

    // ---- per-neighbor precompute ----
    if (tid < NPAD) {
        int j = tid;
        if (j < NNEI) {
            int nj = (a + 1 + j) % A_ATOMS;
            float dx = cx - coords[(c * A_ATOMS + nj) * 3 + 0];
            float dy = cy - coords[(c * A_ATOMS + nj) * 3 + 1];
            float dz = cz - coords[(c * A_ATOMS + nj) * 3 + 2];
            float d  = sqrtf(dx * dx + dy * dy + dz * dz);
            float fcr = (d <= 5.2f) ? 0.5f * cosf(3.14159265358979f * d / 5.2f) + 0.5f : 0.0f;
            float fca = (d <= 3.5f) ? 0.5f * cosf(3.14159265358979f * d / 3.5f) + 0.5f : 0.0f;
            int sp = species[c * A_ATOMS + nj];
            sVD[j]  = make_float4(dx, dy, dz, d);
            sRad[j] = make_float2(d, fcr);
            sFS[j]  = make_float2(fca, __int_as_float(sp));
            sSp[j]  = sp;
        } else { // pad slot 95: contributes exactly zero
            sVD[j]  = make_float4(0.0f, 0.0f, 0.0f, 1e30f);
            sRad[j] = make_float2(1e30f, 0.0f);
            sFS[j]  = make_float2(0.0f, __int_as_float(0));
            sSp[j]  = 0;
        }
    }
    __syncthreads();

    // ---- radial: D(16x16) += maskT(16x4) * rterm(4x16); wave wv owns chunks [wv*6, wv*6+6) ----
    v8f accR = {};
    const float shfr = 0.9f + 0.26875f * (float)l16;   // ShfR[f], f = N = l16
    #pragma unroll
    for (int i = 0; i < CPW; i++) {
        int n0 = (wv * CPW + i) * 4 + k0;              // broadcast reads (same addr per half-wave)
        float2 r0 = sRad[n0];
        float2 r1 = sRad[n0 + 1];
        v2f Af, Bf;
        Af.x = (sSp[n0]     == l16) ? 1.0f : 0.0f;     // A(M=l16, K=k0)
        Af.y = (sSp[n0 + 1] == l16) ? 1.0f : 0.0f;     // A(M=l16, K=k0+1)
        float e0 = r0.x - shfr, e1 = r1.x - shfr;
        Bf.x = 0.25f * __expf(-16.0f * e0 * e0) * r0.y; // B(K=k0,  N=l16)
        Bf.y = 0.25f * __expf(-16.0f * e1 * e1) * r1.y; // B(K=k0+1,N=l16)
        accR = __builtin_amdgcn_wmma_f32_16x16x4_f32(
                   false, Af, false, Bf, (short)0, accR, false, false);
    }

    // ---- angular: two 16x16 accumulators (feature tiles f=0..15, f=16..31) ----
    v8f accA0 = {};
    v8f accA1 = {};
    const int z    = l16 & 7;
    const int sa0  = l16 >> 3;                     // tile0 ShfA idx 0/1; tile1 -> +2
    const float czv = CZ[z], szv = SZ[z];
    const float shfa0 = 0.9f + 0.65f * (float)sa0;
    const float shfa1 = 0.9f + 0.65f * (float)(sa0 + 2);
    const int stageBase = wv * 32;                 // this wave's private staging region

    for (int i = 0; i < BPW; i++) {                // scalar loop: wave wv owns batches [wv*35, wv*35+35)
        int b = wv * BPW + i;
        // phase 1: one pair per lane -> scalars (ca, sin, dmean, fcj, bin q)
        int p = b * 32 + lane;
        float4 ps = make_float4(0.0f, 0.0f, 0.0f, 0.0f);
        int q = 15;                                 // dead row -> zero contribution
        if (p < NPAIR) {
            int i1 = (int)((189.0f - sqrtf(35721.0f - 8.0f * (float)p)) * 0.5f);
            i1 = i1 < 0 ? 0 : (i1 > 93 ? 93 : i1);
            while ((i1 + 1) * (189 - (i1 + 1)) / 2 <= p) i1++;
            while (i1 * (189 - i1) / 2 > p) i1--;
            int i2 = i1 + 1 + (p - i1 * (189 - i1) / 2);
            float4 v1 = sVD[i1];
            float4 v2 = sVD[i2];
            float2 f1 = sFS[i1];
            float2 f2 = sFS[i2];
            float dot = v1.x * v2.x + v1.y * v2.y + v1.z * v2.z;
            float ca  = 0.95f * dot / (v1.w * v2.w);
            ps.x = ca;
            ps.y = sqrtf(fmaxf(0.0f, 1.0f - ca * ca));  // sin(acos(ca)), ca in [-.95,.95]
            ps.z = 0.5f * (v1.w + v2.w);
            ps.w = f1.x * f2.x;
            int s1 = __float_as_int(f1.y), s2 = __float_as_int(f2.y);
            int u = s1 < s2 ? s1 : s2;
            int v = s1 < s2 ? s2 : s1;
            q = u * (7 - u) / 2 + v;                   // unordered species-pair bin 0..9
        }
        sPair[stageBase + lane] = ps;
        sQ[stageBase + lane]    = q;
        __syncthreads();

        // phase 2: 8 K-chunks of 4 pairs, 2 feature tiles each (broadcast LDS reads)
        #pragma unroll
        for (int cc = 0; cc < 8; cc++) {
            int p0 = stageBase + cc * 4 + k0;
            float4 s0 = sPair[p0];
            float4 s1 = sPair[p0 + 1];
            v2f Af, B0, B1;
            Af.x = (sQ[p0]     == l16) ? 1.0f : 0.0f;
            Af.y = (sQ[p0 + 1] == l16) ? 1.0f : 0.0f;
            {   // pair p0 -> K = k0
                float t = 0.5f * (1.0f + s0.x * czv + s0.y * szv);
                float t2 = t * t, t4 = t2 * t2, t8 = t4 * t4, t16 = t8 * t8;
                float g  = 2.0f * (t16 * t16) * s0.w;   // 2 * ((1+cos(th-ShfZ))/2)^32 * fcj
                float e0 = s0.z - shfa0, e1 = s0.z - shfa1;
                B0.x = g * __expf(-8.0f * e0 * e0);
                B1.x = g * __expf(-8.0f * e1 * e1);
            }
            {   // pair p0+1 -> K = k0+1
                float t = 0.5f * (1.0f + s1.x * czv + s1.y * szv);
                float t2 = t * t, t4 = t2 * t2, t8 = t4 * t4, t16 = t8 * t8;
                float g  = 2.0f * (t16 * t16) * s1.w;
                float e0 = s1.z - shfa0, e1 = s1.z - shfa1;
                B0.y = g * __expf(-8.0f * e0 * e0);
                B1.y = g * __expf(-8.0f * e1 * e1);
            }
            accA0 = __builtin_amdgcn_wmma_f32_16x16x4_f32(
                        false, Af, false, B0, (short)0, accA0, false, false);
            accA1 = __builtin_amdgcn_wmma_f32_16x16x4_f32(
                        false, Af, false, B1, (short)0, accA1, false, false);
        }
        __syncthreads();
    }

    // ---- deterministic cross-wave reduction via LDS ----
    // radial: D(M=s, N=f), M = v + 8*half; valid s<4 -> lanes 0-15, v=0..3
    if (half == 0) {
        #pragma unroll
        for (int v = 0; v < 4; v++)
            sRed[wv][v * 16 + l16] = accR[v];
    }
    // angular: M = q = v + 8*half, valid q<10; f = l16 (tile0) / 16+l16 (tile1)
    #pragma unroll
    for (int v = 0; v < 8; v++) {
        int q = v + half * 8;
        if (q < NQ) {
            sRed[wv][64 + q * 32 + l16]      = accA0[v];
            sRed[wv][64 + q * 32 + 16 + l16] = accA1[v];
        }
    }
    __syncthreads();

    float* outAtom = out + (size_t)atomGlobal * FEAT;
    for (int o = tid; o < FEAT; o += 128)
        outAtom[o] = sRed[0][o] + sRed[1][o] + sRed[2][o] + sRed[3][o];
}

extern "C" void kernel_launch(void* const* d_in, const int* in_sizes, int n_in,
                              void* d_out, int out_size, void* d_ws, size_t ws_size,
                              hipStream_t stream) {
    (void)n_in; (void)out_size; (void)d_ws; (void)ws_size;
    const int*   species = (const int*)d_in[0];
    const float* coords  = (const float*)d_in[1];
    float*       out     = (float*)d_out;
    const int numAtoms = in_sizes[0];              // C * 96
    hipLaunchKernelGGL(aev_kernel, dim3(numAtoms), dim3(128), 0, stream,
                       species, coords, out);
}